// perception_attention_mechanism_87497073754204
// MI455X (gfx1250) — compile-verified
//
#include <hip/hip_runtime.h>

#define EPS 1e-8f

typedef __attribute__((ext_vector_type(2))) float v2f;
typedef __attribute__((ext_vector_type(8))) float v8f;

#define M_SLOTS 60          // memory slots (K of the GEMM)
#define CCH     256         // channels
#define HW      3136        // 56*56
#define NTOT    802816      // 256*56*56 (N of the GEMM)
#define BQ      64          // batch (M of the GEMM)

// ---------------------------------------------------------------------------
// Kernel 1: mem_avg[m*256+c] = mean over H*W.  One 128-thread block per row.
// ---------------------------------------------------------------------------
__global__ void __launch_bounds__(128) pool_kernel(const float* __restrict__ mem,
                                                   float* __restrict__ avg) {
  const int row = blockIdx.x;                      // 0 .. 60*256-1
  const float* p = mem + (size_t)row * HW;
  float s = 0.f;
  for (int i = threadIdx.x; i < HW; i += 128) s += p[i];
  __shared__ float sm[128];
  sm[threadIdx.x] = s;
  __syncthreads();
#pragma unroll
  for (int off = 64; off > 0; off >>= 1) {
    if (threadIdx.x < off) sm[threadIdx.x] += sm[threadIdx.x + off];
    __syncthreads();
  }
  if (threadIdx.x == 0) avg[row] = sm[0] * (1.0f / (float)HW);
}

// ---------------------------------------------------------------------------
// Kernel 2: cosine similarity + softmax -> w[64,60]. One block per batch row.
// ---------------------------------------------------------------------------
__global__ void __launch_bounds__(64) weights_kernel(const float* __restrict__ q,
                                                     const float* __restrict__ avg,
                                                     float* __restrict__ w) {
  const int b = blockIdx.x;     // 0..63
  const int t = threadIdx.x;    // 0..63
  __shared__ float qrow[CCH];
  __shared__ float red[64];

  // q-row norm
  float s = 0.f;
  for (int c = t; c < CCH; c += 64) {
    float v = q[b * CCH + c];
    qrow[c] = v;
    s += v * v;
  }
  red[t] = s;
  __syncthreads();
#pragma unroll
  for (int off = 32; off > 0; off >>= 1) {
    if (t < off) red[t] += red[t + off];
    __syncthreads();
  }
  const float qn = fmaxf(sqrtf(red[0]), EPS);
  __syncthreads();

  // per-thread cosine with memory slot t
  float cosv = 0.f;
  if (t < M_SLOTS) {
    const float* mrow = avg + t * CCH;
    float dot = 0.f, mm = 0.f;
    for (int c = 0; c < CCH; ++c) {
      float mv = mrow[c];
      dot += mv * qrow[c];
      mm  += mv * mv;
    }
    cosv = dot / (qn * fmaxf(sqrtf(mm), EPS));
  }

  // softmax over the 60 slots (max then sum, via LDS)
  red[t] = (t < M_SLOTS) ? cosv : -3.0e38f;
  __syncthreads();
#pragma unroll
  for (int off = 32; off > 0; off >>= 1) {
    if (t < off) red[t] = fmaxf(red[t], red[t + off]);
    __syncthreads();
  }
  const float mx = red[0];
  __syncthreads();

  const float e = (t < M_SLOTS) ? __expf(cosv - mx) : 0.f;
  red[t] = e;
  __syncthreads();
#pragma unroll
  for (int off = 32; off > 0; off >>= 1) {
    if (t < off) red[t] += red[t + off];
    __syncthreads();
  }
  if (t < M_SLOTS) w[b * M_SLOTS + t] = e / red[0];
}

// ---------------------------------------------------------------------------
// Kernel 3: out[64, 802816] = W[64,60] @ mem[60, 802816] via f32 WMMA 16x16x4.
// Each wave: one 64(b) x 16(n) tile = 4 accumulators sharing each B operand.
// 8 waves per block -> 128 columns per block; grid = 802816/128 = 6272.
// ---------------------------------------------------------------------------
__global__ void __launch_bounds__(256) wmma_out_kernel(const float* __restrict__ mem,
                                                       const float* __restrict__ wmat,
                                                       float* __restrict__ out) {
  __shared__ float wl[BQ * M_SLOTS];   // 64x60 weights, 15360 B
  for (int i = threadIdx.x; i < BQ * M_SLOTS; i += 256) wl[i] = wmat[i];
  __syncthreads();

  const int lane = threadIdx.x & 31;
  const int wave = threadIdx.x >> 5;   // 0..7
  const int l    = lane & 15;
  const int hi   = lane >> 4;          // lane-group select (0 or 1)
  const int n    = blockIdx.x * 128 + wave * 16 + l;

  v8f acc0 = {}, acc1 = {}, acc2 = {}, acc3 = {};

#pragma unroll
  for (int kc = 0; kc < 15; ++kc) {
    // Fold the lane-group K offset in: lanes 0-15 use K={k,k+1}, 16-31 use K+2.
    const int k = kc * 4 + 2 * hi;

    // B (4x16 tile of memory_pool), streamed, read exactly once per element.
    v2f bm;
    bm.x = __builtin_nontemporal_load(&mem[(k)     * NTOT + n]);
    bm.y = __builtin_nontemporal_load(&mem[(k + 1) * NTOT + n]);

    // A (16x4 tiles of W) from LDS, one per b-tile.
    v2f a0, a1, a2, a3;
    a0.x = wl[( 0 + l) * M_SLOTS + k];  a0.y = wl[( 0 + l) * M_SLOTS + k + 1];
    a1.x = wl[(16 + l) * M_SLOTS + k];  a1.y = wl[(16 + l) * M_SLOTS + k + 1];
    a2.x = wl[(32 + l) * M_SLOTS + k];  a2.y = wl[(32 + l) * M_SLOTS + k + 1];
    a3.x = wl[(48 + l) * M_SLOTS + k];  a3.y = wl[(48 + l) * M_SLOTS + k + 1];

    acc0 = __builtin_amdgcn_wmma_f32_16x16x4_f32(false, a0, false, bm, (short)0, acc0, false, false);
    acc1 = __builtin_amdgcn_wmma_f32_16x16x4_f32(false, a1, false, bm, (short)0, acc1, false, false);
    acc2 = __builtin_amdgcn_wmma_f32_16x16x4_f32(false, a2, false, bm, (short)0, acc2, false, false);
    acc3 = __builtin_amdgcn_wmma_f32_16x16x4_f32(false, a3, false, bm, (short)0, acc3, false, false);
  }

  // C/D layout: VGPR r -> row r (lanes 0-15) / row r+8 (lanes 16-31).
#pragma unroll
  for (int r = 0; r < 8; ++r) {
    const int mrow = r + 8 * hi;
    __builtin_nontemporal_store(acc0[r], &out[( 0 + mrow) * NTOT + n]);
    __builtin_nontemporal_store(acc1[r], &out[(16 + mrow) * NTOT + n]);
    __builtin_nontemporal_store(acc2[r], &out[(32 + mrow) * NTOT + n]);
    __builtin_nontemporal_store(acc3[r], &out[(48 + mrow) * NTOT + n]);
  }
}

// ---------------------------------------------------------------------------
extern "C" void kernel_launch(void* const* d_in, const int* in_sizes, int n_in,
                              void* d_out, int out_size, void* d_ws, size_t ws_size,
                              hipStream_t stream) {
  (void)in_sizes; (void)n_in; (void)out_size; (void)ws_size;

  const float* mem = (const float*)d_in[0];   // [60,256,56,56]
  const float* q   = (const float*)d_in[1];   // [64,256]
  float* out = (float*)d_out;                 // [64,256,56,56]

  float* avg = (float*)d_ws;                  // 60*256 floats
  float* w   = avg + M_SLOTS * CCH;           // 64*60 floats

  pool_kernel<<<M_SLOTS * CCH, 128, 0, stream>>>(mem, avg);
  weights_kernel<<<BQ, 64, 0, stream>>>(q, avg, w);
  wmma_out_kernel<<<NTOT / 128, 256, 0, stream>>>(mem, w, out);
}